// SpeckleDistribution_2267742732877
// MI455X (gfx1250) — compile-verified
//
#include <hip/hip_runtime.h>

// dist[b,p] = lambda15 * [expm(x[b,p] * theta)]_{0,15}
// theta = diag(-lambda) + superdiag(lambda[:-1]), lambda = lmbda_sqrt^2.
// One wave32 per (b,p): scaling-and-squaring expm of a 16x16 f32 matrix
// built from chained V_WMMA_F32_16X16X4_F32, with operand re-layout staged
// through a bank-conflict-free padded LDS tile (row stride 20 floats).

typedef __attribute__((ext_vector_type(2))) float v2f;
typedef __attribute__((ext_vector_type(8))) float v8f;

#define LSTRIDE 20          // floats per row: 80B -> rows 16B aligned, banks conflict-free
#define NTAY    10          // Taylor degree for ||S|| <= 0.5 : err ~ 0.5^11/11! ~ 1e-11

// One 16x16 x 16x16 f32 matmul = 4 chained WMMA 16x16x4 over K.
// a[k] holds A-format chunk k (lane<16: A[row][4k..4k+1], lane>=16: A[row][4k+2..4k+3]).
// B-format chunk k: vgpr0 = B[4k+2*hi][col], vgpr1 = B[4k+1+2*hi][col].
__device__ __forceinline__ v8f mm16x16(const v2f a[4], const float* Bt, int col, int hi) {
  v8f c = {};
  #pragma unroll
  for (int k = 0; k < 4; ++k) {
    const int rb = 4 * k + 2 * hi;
    v2f b;
    b.x = Bt[rb * LSTRIDE + col];
    b.y = Bt[(rb + 1) * LSTRIDE + col];
    // (neg_a, A, neg_b, B, c_mod, C, reuse_a, reuse_b)
    c = __builtin_amdgcn_wmma_f32_16x16x4_f32(false, a[k], false, b, (short)0, c,
                                              false, false);
  }
  return c;
}

__global__ __launch_bounds__(256) void speckle_expm_kernel(
    const float* __restrict__ x, const float* __restrict__ lam_sq,
    float* __restrict__ out, int n) {
  __shared__ float shS[8][16 * LSTRIDE];  // scaled generator S = (t/2^s) * theta
  __shared__ float shT[8][16 * LSTRIDE];  // working matrix T

  const int lane = threadIdx.x & 31;
  const int wv   = __builtin_amdgcn_readfirstlane((int)(threadIdx.x >> 5));
  const int idx  = blockIdx.x * 8 + wv;
  if (idx >= n) return;  // wave-uniform: EXEC stays all-ones for WMMA

  float* S = shS[wv];
  float* T = shT[wv];
  const int col = lane & 15;  // also "row" for A-operand / row-build duty
  const int hi  = lane >> 4;

  const float t = x[idx];  // wave-uniform -> scalar load

  // lambda_i = lmbda_sqrt[i]^2 ; ||t*theta||_inf = t * max(2*l_0..14, l_15)
  float lmax = 0.f, lam15 = 0.f;
  #pragma unroll
  for (int i = 0; i < 16; ++i) {
    const float ls = lam_sq[i];
    const float l  = ls * ls;
    if (i < 15) lmax = fmaxf(lmax, 2.f * l);
    else { lmax = fmaxf(lmax, l); lam15 = l; }
  }
  const float norm = t * lmax;
  int sqr = 0;
  if (norm > 0.5f) sqr = (int)ceilf(log2f(norm * 2.0f));  // ||S|| = norm/2^sqr <= 0.5
  sqr = (sqr < 0) ? 0 : ((sqr > 30) ? 30 : sqr);
  const float r = ldexpf(t, -sqr);  // S = r * theta

  // Build S (bidiagonal) and T0 = I + S/NTAY. Lane l (<16) writes row l.
  const float lsr  = lam_sq[col];
  const float lrow = lsr * lsr;
  if (hi == 0) {
    #pragma unroll
    for (int c = 0; c < 16; ++c) {
      const float vs = (c == col) ? (-r * lrow)
                     : ((c == col + 1) ? (r * lrow) : 0.f);
      S[col * LSTRIDE + c] = vs;
      T[col * LSTRIDE + c] = vs * (1.f / (float)NTAY) + ((c == col) ? 1.f : 0.f);
    }
  }
  asm volatile("s_wait_dscnt 0" ::: "memory");  // cross-lane LDS handoff

  // A-format of S stays in registers for the whole Horner phase (S is constant).
  v2f aS[4];
  #pragma unroll
  for (int k = 0; k < 4; ++k)
    aS[k] = *(const v2f*)&S[col * LSTRIDE + 4 * k + 2 * hi];  // 8B aligned -> ds_load_b64

  // Horner: T <- I + (S*T)/j for j = NTAY-1 .. 1  => T = sum_{k<=NTAY} S^k/k!
  // Fully unrolled: constant trip count makes every 1/j a compile-time literal
  // (bit-identical to the IEEE division result), eliminating v_div_* chains.
  float d[8];
  #pragma unroll
  for (int j = NTAY - 1; j >= 1; --j) {
    const v8f p     = mm16x16(aS, T, col, hi);
    const float inv = 1.f / (float)j;  // compile-time constant after unroll
    #pragma unroll
    for (int q = 0; q < 8; ++q) {
      const int rw = q + 8 * hi;
      d[q] = p[q] * inv + ((rw == col) ? 1.f : 0.f);
    }
    asm volatile("" ::: "memory");  // keep T-reads above the overwrite below
    #pragma unroll
    for (int q = 0; q < 8; ++q)
      T[(q + 8 * hi) * LSTRIDE + col] = d[q];
    asm volatile("s_wait_dscnt 0" ::: "memory");
  }

  // Squaring phase: E <- E*E, sqr times (wave-uniform trip count).
  for (int i = 0; i < sqr; ++i) {
    v2f aT[4];
    #pragma unroll
    for (int k = 0; k < 4; ++k)
      aT[k] = *(const v2f*)&T[col * LSTRIDE + 4 * k + 2 * hi];
    const v8f p = mm16x16(aT, T, col, hi);
    #pragma unroll
    for (int q = 0; q < 8; ++q) d[q] = p[q];
    asm volatile("" ::: "memory");
    #pragma unroll
    for (int q = 0; q < 8; ++q)
      T[(q + 8 * hi) * LSTRIDE + col] = d[q];
    asm volatile("s_wait_dscnt 0" ::: "memory");
  }

  // E[0][15] lives in C/D vgpr0 at lane 15 (row 0, col 15).
  if (lane == 15) out[idx] = lam15 * d[0];
}

extern "C" void kernel_launch(void* const* d_in, const int* in_sizes, int n_in,
                              void* d_out, int out_size, void* d_ws, size_t ws_size,
                              hipStream_t stream) {
  (void)n_in; (void)out_size; (void)d_ws; (void)ws_size;
  const float* x   = (const float*)d_in[0];   // (512*512,) f32
  const float* lsq = (const float*)d_in[1];   // (16,) f32
  float* out = (float*)d_out;                 // (512*512,) f32
  const int n = in_sizes[0];
  const int waves_per_block = 8;              // 256 threads = 8 wave32
  const int blocks = (n + waves_per_block - 1) / waves_per_block;
  speckle_expm_kernel<<<blocks, 256, 0, stream>>>(x, lsq, out, n);
}